// ClassBalancedSegmentationLosses_30545807409282
// MI455X (gfx1250) — compile-verified
//
#include <hip/hip_runtime.h>
#include <math.h>

#define NCLASS 19
#define HW (512 * 512)          // spatial size per image (2^18), matches reference
#define LOG2E 1.4426950408889634f
#define LN2   0.6931471805599453f

static constexpr float BETA = 1.0f - 0.001f;

// fast hardware transcendentals: v_exp_f32 (2^x) and v_log_f32 (log2 x)
__device__ __forceinline__ float fexp2(float x) { return __builtin_amdgcn_exp2f(x); }
__device__ __forceinline__ float flog2(float x) { return __builtin_amdgcn_logf(x); }

// ---------------- workspace layout (in d_ws) ----------------
// u32  [0  .. 18]  : class counts
// f32  [32 .. 50]  : class weights w[c]
// f32  [64]        : numerator   sum(pix_w * nll)
// f32  [65]        : denominator sum(pix_w)

__global__ void cbsl_zero_kernel(unsigned* ws) {
    int i = threadIdx.x;
    if (i < 96) ws[i] = 0u;
}

// ---------------- histogram of target over 19 classes ----------------
__global__ void cbsl_hist_kernel(const int4* __restrict__ tgt4, int n4,
                                 unsigned* __restrict__ counts) {
    __shared__ unsigned bins[32];           // padded; 19 used
    int tid = threadIdx.x;
    if (tid < 32) bins[tid] = 0u;
    __syncthreads();

    int stride = gridDim.x * blockDim.x;
    for (int i = blockIdx.x * blockDim.x + tid; i < n4; i += stride) {
        // CDNA5 prefetch path: global_prefetch_b8 for the next grid-stride chunk
        __builtin_prefetch((const void*)(tgt4 + i + stride), 0, 1);
        int4 t = tgt4[i];
        if (t.x >= 0 && t.x < NCLASS) atomicAdd(&bins[t.x], 1u);
        if (t.y >= 0 && t.y < NCLASS) atomicAdd(&bins[t.y], 1u);
        if (t.z >= 0 && t.z < NCLASS) atomicAdd(&bins[t.z], 1u);
        if (t.w >= 0 && t.w < NCLASS) atomicAdd(&bins[t.w], 1u);
    }
    __syncthreads();
    if (tid < NCLASS) atomicAdd(&counts[tid], bins[tid]);
}

// ---------------- class-balanced weights (one wave32) ----------------
__global__ void cbsl_weight_kernel(const unsigned* __restrict__ counts,
                                   float* __restrict__ w) {
    int c = threadIdx.x;
    if (c < NCLASS) {
        unsigned n = counts[c];
        float fn = (float)n;
        // beta^n = exp2(n * log2(beta)); underflows cleanly to 0 for large n
        float bp = fexp2(fn * flog2(BETA));
        float ww = (1.0f - BETA) / (1.0f - bp);
        w[c] = (n > 0u) ? ww : 0.0f;
    }
}

// per-subpixel tail: weight gather + nll; xt reload is a guaranteed L0 hit
__device__ __forceinline__ void cbsl_accum(int t, float lse, const float* base, int off,
                                           const float* sw, float& num, float& den) {
    bool valid = (t >= 0) && (t < NCLASS);
    int tc = valid ? t : 0;
    float xt = base[tc * HW + off];     // line just loaded by this lane -> WGP$ hit
    float pw = valid ? sw[tc] : 0.0f;
    num += pw * (lse - xt);             // pw==0 kills invalid pixels
    den += pw;
}

// ---------------- weighted NLL over log_softmax, fused reduction ----------------
// one float4 (4 consecutive pixels) per thread: 19x global_load_b128
__global__ void __launch_bounds__(256)
cbsl_loss_kernel(const float* __restrict__ pred,
                 const int* __restrict__ tgt,
                 const float* __restrict__ wcls,
                 float* __restrict__ acc,        // acc[0]=num, acc[1]=den
                 int nquad) {
    __shared__ float sw[32];
    if (threadIdx.x < NCLASS) sw[threadIdx.x] = wcls[threadIdx.x];
    __syncthreads();

    int q = blockIdx.x * blockDim.x + threadIdx.x;
    float num = 0.0f, den = 0.0f;

    if (q < nquad) {
        int p = q << 2;                     // first pixel of the quad
        int b = p >> 18;                    // p / HW  (HW = 2^18)
        int s = p & (HW - 1);               // p % HW  (4-aligned; never crosses a plane)
        const float* base = pred + b * (NCLASS * HW) + s;
        int4 t = ((const int4*)tgt)[q];

        float4 xs[NCLASS];
        float mx = -INFINITY, my = -INFINITY, mz = -INFINITY, mw = -INFINITY;
#pragma unroll
        for (int c = 0; c < NCLASS; ++c) {
            float4 x = *(const float4*)(base + c * HW);   // global_load_b128, coalesced
            xs[c] = x;
            mx = fmaxf(mx, x.x);
            my = fmaxf(my, x.y);
            mz = fmaxf(mz, x.z);
            mw = fmaxf(mw, x.w);
        }

        float mlx = mx * LOG2E, mly = my * LOG2E, mlz = mz * LOG2E, mlw = mw * LOG2E;
        float sx = 0.0f, sy = 0.0f, sz = 0.0f, s_w = 0.0f;
#pragma unroll
        for (int c = 0; c < NCLASS; ++c) {
            sx  += fexp2(fmaf(xs[c].x, LOG2E, -mlx));     // v_fma + v_exp_f32
            sy  += fexp2(fmaf(xs[c].y, LOG2E, -mly));
            sz  += fexp2(fmaf(xs[c].z, LOG2E, -mlz));
            s_w += fexp2(fmaf(xs[c].w, LOG2E, -mlw));
        }
        float lsex = mx + flog2(sx)  * LN2;
        float lsey = my + flog2(sy)  * LN2;
        float lsez = mz + flog2(sz)  * LN2;
        float lsew = mw + flog2(s_w) * LN2;

        cbsl_accum(t.x, lsex, base, 0, sw, num, den);
        cbsl_accum(t.y, lsey, base, 1, sw, num, den);
        cbsl_accum(t.z, lsez, base, 2, sw, num, den);
        cbsl_accum(t.w, lsew, base, 3, sw, num, den);
    }

    // wave32 reduction (gfx1250 is wave32-only: width MUST be 32)
#pragma unroll
    for (int off = 16; off > 0; off >>= 1) {
        num += __shfl_down(num, off, 32);
        den += __shfl_down(den, off, 32);
    }

    __shared__ float snum[8];
    __shared__ float sden[8];
    int lane = threadIdx.x & 31;
    int wid  = threadIdx.x >> 5;
    if (lane == 0) { snum[wid] = num; sden[wid] = den; }
    __syncthreads();

    if (wid == 0) {
        num = (lane < 8) ? snum[lane] : 0.0f;
        den = (lane < 8) ? sden[lane] : 0.0f;
#pragma unroll
        for (int off = 4; off > 0; off >>= 1) {
            num += __shfl_down(num, off, 32);
            den += __shfl_down(den, off, 32);
        }
        if (lane == 0) {
            atomicAdd(&acc[0], num);        // global_atomic_add_f32
            atomicAdd(&acc[1], den);
        }
    }
}

__global__ void cbsl_finalize_kernel(const float* __restrict__ acc,
                                     float* __restrict__ out) {
    if (threadIdx.x == 0) out[0] = acc[0] / acc[1];
}

extern "C" void kernel_launch(void* const* d_in, const int* in_sizes, int n_in,
                              void* d_out, int out_size, void* d_ws, size_t ws_size,
                              hipStream_t stream) {
    const float* pred = (const float*)d_in[0];  // [8, 19, 512, 512] f32
    const int*   tgt  = (const int*)d_in[1];    // [8, 512, 512] i32
    float*       out  = (float*)d_out;          // scalar f32

    int npix = in_sizes[1];                     // 8*512*512 = 2,097,152

    unsigned* counts = (unsigned*)d_ws;             // 19 u32
    float*    wcls   = (float*)d_ws + 32;           // 19 f32
    float*    acc    = (float*)d_ws + 64;           // 2  f32

    // 1) zero workspace (harness poisons ws; must re-zero every call)
    cbsl_zero_kernel<<<1, 128, 0, stream>>>((unsigned*)d_ws);

    // 2) histogram (int4 loads, LDS bins, prefetch)
    int n4 = npix >> 2;
    cbsl_hist_kernel<<<1024, 256, 0, stream>>>((const int4*)tgt, n4, counts);

    // 3) class-balanced weights (one wave)
    cbsl_weight_kernel<<<1, 32, 0, stream>>>(counts, wcls);

    // 4) fused log-softmax + weighted NLL + global reduction (4 px / thread)
    int nquad = npix >> 2;
    int blocks = (nquad + 255) / 256;
    cbsl_loss_kernel<<<blocks, 256, 0, stream>>>(pred, tgt, wcls, acc, nquad);

    // 5) finalize
    cbsl_finalize_kernel<<<1, 32, 0, stream>>>(acc, out);
}